// HashEncoderTaichi_26345329393885
// MI455X (gfx1250) — compile-verified
//
#include <hip/hip_runtime.h>
#include <math.h>

// ---------------------------------------------------------------------------
// Instant-NGP multiresolution hash encoding, MI455X (gfx1250, wave32).
//
// Layout: 256 threads/block = 8 waves. threadIdx.x & 15 = level (0..15),
// threadIdx.x >> 4 = point slot -> each wave handles 2 points x 16 levels and
// its 32 float2 stores form one contiguous 256B burst (coalesced output).
//
// Levels 0..1 (dense, 14744 entries, 117,952 B) are staged into LDS once per
// block via the Tensor Data Mover (TENSOR_LOAD_TO_LDS + s_wait_tensorcnt),
// serving the hottest 12.5% of gathers from LDS instead of L2. The remaining
// table stays resident in the 192MB L2. Output is written with nontemporal
// stores so the 128MB stream does not evict the table.
// ---------------------------------------------------------------------------

#define NLEVELS 16
#define POINTS_PER_ITER 16u   // points per block iteration (256 threads / 16 levels)

struct LevelMeta {
  float    scale[NLEVELS];
  unsigned offset[NLEVELS];  // in float2 entries
  unsigned size[NLEVELS];    // in float2 entries
  unsigned mask[NLEVELS];    // size-1 for hashed (pow2), 0xFFFFFFFF for dense
  unsigned res[NLEVELS];
};

typedef unsigned int u32x4 __attribute__((ext_vector_type(4)));
typedef int          i32x8 __attribute__((ext_vector_type(8)));
typedef int          i32x4 __attribute__((ext_vector_type(4)));

__global__ __launch_bounds__(256) void hash_encode_kernel(
    const float* __restrict__ positions,
    const float* __restrict__ table,
    float* __restrict__ out,
    LevelMeta meta, unsigned npoints, unsigned lds_entries)
{
  extern __shared__ float lds_tab[];   // lds_entries float2's

  // ---- Stage levels [0, lds_entries) into LDS ----
#if __has_builtin(__builtin_amdgcn_tensor_load_to_lds) && __has_builtin(__builtin_amdgcn_s_wait_tensorcnt)
  if ((threadIdx.x >> 5) == 0) {   // one wave issues the TDM descriptor
    unsigned long long ga = (unsigned long long)table;
    unsigned ldsOff = (unsigned)__builtin_amdgcn_groupstaticsize(); // dyn-LDS base
    unsigned units  = lds_entries;   // tile length in 8-byte (data_size=3) units
    // D# group 0: count=1 | lds_addr | global_addr[56:0] | type=2
    u32x4 g0;
    g0.x = 1u;
    g0.y = ldsOff;
    g0.z = (unsigned)(ga & 0xFFFFFFFFu);
    g0.w = (unsigned)((ga >> 32) & 0x01FFFFFFu) | (2u << 30);
    // D# group 1: data_size=8B, tensor_dim0=units, tensor_dim1=1,
    //             tile_dim0=units (1-D tile), dim0_stride=units
    i32x8 g1;
    g1[0] = (int)(3u << 16);                                       // data_size
    g1[1] = (int)((units & 0xFFFFu) << 16);                        // tdim0 lo16
    g1[2] = (int)(((units >> 16) & 0xFFFFu) | (1u << 16));         // tdim0 hi | tdim1 lo
    g1[3] = (int)((units & 0xFFFFu) << 16);                        // tile_dim0
    g1[4] = 0;                                                     // tile_dim1/2 = 0
    g1[5] = (int)units;                                            // dim0_stride lo32
    g1[6] = 0;
    g1[7] = 0;
    i32x4 g2; g2[0] = 1; g2[1] = 1; g2[2] = 0; g2[3] = 0;          // tdim2/3=1
    i32x4 g3; g3[0] = 0; g3[1] = (int)(1u << 16); g3[2] = 0; g3[3] = 0; // tdim4=1
    i32x8 g4 = {0, 0, 0, 0, 0, 0, 0, 0};                           // unused (6-arg form)
    __builtin_amdgcn_tensor_load_to_lds(g0, g1, g2, g3, g4, 0);
    __builtin_amdgcn_s_wait_tensorcnt(0);
  }
#else
  {   // fallback: cooperative b128 copy (global_load_b128 + ds_store_b128)
    const uint4* src = (const uint4*)table;
    uint4*       dst = (uint4*)lds_tab;
    unsigned nchunks = lds_entries >> 1;   // entries*8B / 16B
    for (unsigned i = threadIdx.x; i < nchunks; i += 256u) dst[i] = src[i];
  }
#endif
  __syncthreads();

  const unsigned level = threadIdx.x & 15u;
  const unsigned slot  = threadIdx.x >> 4;

  const float    scale = meta.scale[level];
  const unsigned sz    = meta.size[level];
  const unsigned msk   = meta.mask[level];
  const unsigned off   = meta.offset[level];
  const unsigned res   = meta.res[level];
  const bool     dense = (msk == 0xFFFFFFFFu);
  const bool     use_lds = (off + sz) <= lds_entries;
  const unsigned res2  = res * res;

  const float2* __restrict__ gtab = (const float2*)table;
  const float2* ltab = (const float2*)lds_tab;

  for (unsigned p = blockIdx.x * POINTS_PER_ITER + slot; p < npoints;
       p += gridDim.x * POINTS_PER_ITER) {
    const float* pp = positions + 3ull * p;
    // x01 = (pos+1)*0.5 ; pos01 = x01*scale + 0.5
    float x = (pp[0] + 1.0f) * 0.5f * scale + 0.5f;
    float y = (pp[1] + 1.0f) * 0.5f * scale + 0.5f;
    float z = (pp[2] + 1.0f) * 0.5f * scale + 0.5f;
    float fx = floorf(x), fy = floorf(y), fz = floorf(z);
    float tx = x - fx, ty = y - fy, tz = z - fz;
    unsigned gx = (unsigned)fx, gy = (unsigned)fy, gz = (unsigned)fz;
    float wx[2] = {1.0f - tx, tx};
    float wy[2] = {1.0f - ty, ty};
    float wz[2] = {1.0f - tz, tz};

    // 8 corner indices: dense = linear index, hashed = NGP fast_hash.
    // No integer division: hashed sizes are 2^19 (mask); dense h < 2*size
    // always, so modulo == single conditional subtract.
    unsigned idx[8];
#pragma unroll
    for (int c = 0; c < 8; ++c) {
      unsigned sx = c & 1, sy = (c >> 1) & 1, szl = (c >> 2) & 1;
      unsigned cx = gx + sx, cy = gy + sy, cz = gz + szl;
      unsigned hd = cx + cy * res + cz * res2;
      unsigned hh = cx ^ (cy * 2654435761u) ^ (cz * 805459861u);
      unsigned h  = dense ? hd : hh;
      h &= msk;
      h = (h >= sz) ? (h - sz) : h;
      idx[c] = h + off;
    }

    float a0 = 0.0f, a1 = 0.0f;
    if (use_lds) {   // lanes of levels 0..1: gather from LDS (ds_load_b64)
#pragma unroll
      for (int c = 0; c < 8; ++c) {
        float2 f = ltab[idx[c]];
        float  w = wx[c & 1] * wy[(c >> 1) & 1] * wz[(c >> 2) & 1];
        a0 = fmaf(w, f.x, a0);
        a1 = fmaf(w, f.y, a1);
      }
    } else {          // lanes of levels 2..15: gather from L2-resident table
#pragma unroll
      for (int c = 0; c < 8; ++c) {
        float2 f = gtab[idx[c]];
        float  w = wx[c & 1] * wy[(c >> 1) & 1] * wz[(c >> 2) & 1];
        a0 = fmaf(w, f.x, a0);
        a1 = fmaf(w, f.y, a1);
      }
    }

    // Coalesced nontemporal b64 store: wave covers 256B contiguous.
    float2 r; r.x = a0; r.y = a1;
    unsigned long long bits;
    __builtin_memcpy(&bits, &r, 8);
    __builtin_nontemporal_store(bits,
        (unsigned long long*)(out + 32ull * p + 2u * level));
  }
}

extern "C" void kernel_launch(void* const* d_in, const int* in_sizes, int n_in,
                              void* d_out, int out_size, void* d_ws, size_t ws_size,
                              hipStream_t stream) {
  (void)n_in; (void)out_size; (void)d_ws; (void)ws_size;
  const float* positions = (const float*)d_in[0];
  const float* table     = (const float*)d_in[1];
  float*       out       = (float*)d_out;
  unsigned npoints = (unsigned)(in_sizes[0] / 3);

  // Level metadata, same double-precision formulas as the reference.
  LevelMeta meta;
  unsigned off = 0, lds_entries = 0;
  const double logB = log(1.3195079565048218);
  for (int i = 0; i < NLEVELS; ++i) {
    double s = 16.0 * exp((double)i * logB) - 1.0;
    unsigned res = (unsigned)ceil(s) + 1u;
    unsigned long long p3 = (unsigned long long)res * res * res;
    unsigned long long p  = (p3 % 8ull) ? ((p3 + 7ull) / 8ull) * 8ull : p3;
    if (p > 524288ull) p = 524288ull;           // MAX_PARAMS = 2^19
    bool dense = (p3 <= p);
    meta.scale[i]  = (float)s;
    meta.offset[i] = off;
    meta.size[i]   = (unsigned)p;
    meta.mask[i]   = dense ? 0xFFFFFFFFu : (unsigned)(p - 1ull); // hashed sizes are 2^19
    meta.res[i]    = res;
    off += (unsigned)p;
    if (i < 2) lds_entries = off;               // stage levels 0..1 in LDS
  }

  size_t ldsBytes = (size_t)lds_entries * 8;    // 117,952 B (< 320KB/WGP)
  (void)hipFuncSetAttribute(reinterpret_cast<const void*>(hash_encode_kernel),
                            hipFuncAttributeMaxDynamicSharedMemorySize,
                            (int)ldsBytes);

  unsigned groups = (npoints + POINTS_PER_ITER - 1) / POINTS_PER_ITER;
  unsigned blocks = groups < 2048u ? groups : 2048u;
  hash_encode_kernel<<<blocks, 256, ldsBytes, stream>>>(
      positions, table, out, meta, npoints, lds_entries);
}